// InternalAligner_6356551598476
// MI455X (gfx1250) — compile-verified
//
#include <hip/hip_runtime.h>
#include <hip/hip_bf16.h>

#define Bn 32
#define Sn 512
#define Tn 2048
#define DTXT 256
#define DMEL 80

typedef float v2f __attribute__((ext_vector_type(2)));
typedef float v8f __attribute__((ext_vector_type(8)));

// D = A(16x4,f32) * B(4x16,f32) + C(16x16,f32) -- full-precision WMMA (CDNA5)
__device__ __forceinline__ v8f wmma4(v2f a, v2f b, v8f c) {
  return __builtin_amdgcn_wmma_f32_16x16x4_f32(false, a, false, b, (short)0, c,
                                               false, false);
}

// Async global->LDS 16B copy (gfx1250), GVS addressing: saddr base + vaddr off.
__device__ __forceinline__ void async_g2l_b128(unsigned lds_off, unsigned goff,
                                               const float* base) {
  asm volatile("global_load_async_to_lds_b128 %0, %1, %2"
               :: "v"(lds_off), "v"(goff), "s"(base) : "memory");
}
__device__ __forceinline__ void wait_asynccnt0() {
  asm volatile("s_wait_asynccnt 0x0" ::: "memory");
}

// ---------------------------------------------------------------------------
// Repack K=3 conv weights (Cout,Cin,3) -> pair-interleaved B-operand layout
// Wp[k>>1][n][2] so a WMMA B-fragment is one aligned float2 load.
//   tw1: K=768,N=512   mw1: K=240,N=160   mw2: K=480,N=80
// ---------------------------------------------------------------------------
__global__ void k_repack(const float* __restrict__ tw1, const float* __restrict__ mw1,
                         const float* __restrict__ mw2, float* __restrict__ W1p,
                         float* __restrict__ W1m, float* __restrict__ W2m) {
  const int i = blockIdx.x * blockDim.x + threadIdx.x;
  const int n1 = 384 * 512, n2 = 120 * 160, n3 = 240 * 80;
  if (i < n1) {
    const int k2 = i / 512, n = i % 512, k = k2 * 2;
    W1p[(size_t)k2 * 1024 + n * 2 + 0] = tw1[(size_t)n * 768 + (k & 255) * 3 + (k >> 8)];
    W1p[(size_t)k2 * 1024 + n * 2 + 1] = tw1[(size_t)n * 768 + ((k + 1) & 255) * 3 + ((k + 1) >> 8)];
  } else if (i < n1 + n2) {
    const int j = i - n1, k2 = j / 160, n = j % 160, k = k2 * 2;
    W1m[(size_t)k2 * 320 + n * 2 + 0] = mw1[(size_t)n * 240 + (k % 80) * 3 + (k / 80)];
    W1m[(size_t)k2 * 320 + n * 2 + 1] = mw1[(size_t)n * 240 + ((k + 1) % 80) * 3 + ((k + 1) / 80)];
  } else if (i < n1 + n2 + n3) {
    const int j = i - n1 - n2, k2 = j / 80, n = j % 80, k = k2 * 2;
    W2m[(size_t)k2 * 160 + n * 2 + 0] = mw2[(size_t)n * 480 + (k % 160) * 3 + (k / 160)];
    W2m[(size_t)k2 * 160 + n * 2 + 1] = mw2[(size_t)n * 480 + ((k + 1) % 160) * 3 + ((k + 1) / 160)];
  }
}

// ---------------------------------------------------------------------------
// Text encoder: h = relu(conv1d(x,K=3,256->512)); emb = conv1d(h,K=1,512->80)
// Stage 1: one A-fragment feeds 4 accumulators (4 n-tiles per wave).
// ---------------------------------------------------------------------------
__global__ void k_text_enc(const float* __restrict__ text,
                           const float* __restrict__ W1p, const float* __restrict__ tb1,
                           const float* __restrict__ tw2, const float* __restrict__ tb2,
                           const int* __restrict__ src_len,
                           float* __restrict__ text_emb) {
  __shared__ float h[16 * 512];
  const int b = blockIdx.y, s0 = blockIdx.x * 16;
  const int tid = threadIdx.x, w = tid >> 5, lane = tid & 31;
  const int m = lane & 15, half = lane >> 4;
  const float* xb = text + (size_t)b * Sn * DTXT;

  // Stage 1: h(16x512) = relu(im2col(16x768) @ W1'(768x512) + b1)
  v8f acc[4] = {};
  for (int kb = 0; kb < 768; kb += 4) {
    const int col = kb + 2 * half;
    const int seq = s0 + m - 1 + (col >> 8);
    const int ci = col & 255;
    v2f a = {};
    if (seq >= 0 && seq < Sn) a = *(const v2f*)(xb + (size_t)seq * DTXT + ci);
    const float* wp = W1p + (size_t)(col >> 1) * 1024 + (w * 64 + m) * 2;
#pragma unroll
    for (int i = 0; i < 4; ++i) {
      v2f bf = *(const v2f*)(wp + i * 32);  // +16 columns
      acc[i] = wmma4(a, bf, acc[i]);
    }
  }
#pragma unroll
  for (int i = 0; i < 4; ++i) {
    const int n = w * 64 + i * 16 + m;
    const float bias = tb1[n];
    for (int r = 0; r < 8; ++r) {
      const int mm = r + 8 * half;
      float v = acc[i][r] + bias;
      h[mm * 512 + n] = v > 0.f ? v : 0.f;
    }
  }
  __syncthreads();

  // Stage 2: emb(16x80) = h(16x512) @ W2^T(512x80) + b2, masked by src_len
  if (w < 5) {
    const int co = w * 16 + m;
    v8f acc2 = {};
    for (int kb = 0; kb < 512; kb += 4) {
      v2f a = *(const v2f*)(&h[m * 512 + kb + 2 * half]);
      v2f bf = *(const v2f*)(tw2 + (size_t)co * 512 + kb + 2 * half);
      acc2 = wmma4(a, bf, acc2);
    }
    const float bias = tb2[co];
    const int slen = src_len[b];
    for (int r = 0; r < 8; ++r) {
      const int s = s0 + r + 8 * half;
      const float v = (s < slen) ? acc2[r] + bias : 0.f;
      text_emb[((size_t)b * Sn + s) * DMEL + co] = v;
    }
  }
}

// ---------------------------------------------------------------------------
// Mel conv1: h1 = leaky_relu(conv1d(mel,K=3,80->160)), 16 t-rows per block.
// ---------------------------------------------------------------------------
__global__ void k_mel_conv1(const float* __restrict__ mel,
                            const float* __restrict__ W1m, const float* __restrict__ mb1,
                            float* __restrict__ h1) {
  const int b = blockIdx.y, t0 = blockIdx.x * 16;
  const int tid = threadIdx.x, w = tid >> 5, lane = tid & 31;
  const int m = lane & 15, half = lane >> 4;
  const float* xb = mel + (size_t)b * Tn * DMEL;

  for (int tile = w; tile < 10; tile += 8) {
    const int n = tile * 16 + m;
    v8f acc = {};
    for (int kb = 0; kb < 240; kb += 4) {
      const int col = kb + 2 * half;
      const int so = (col >= 160) ? 2 : ((col >= 80) ? 1 : 0);
      const int ci = col - so * 80;
      const int seq = t0 + m - 1 + so;
      v2f a = {};
      if (seq >= 0 && seq < Tn) a = *(const v2f*)(xb + (size_t)seq * DMEL + ci);
      v2f bf = *(const v2f*)(W1m + (size_t)(col >> 1) * 320 + n * 2);
      acc = wmma4(a, bf, acc);
    }
    const float bias = mb1[n];
    for (int r = 0; r < 8; ++r) {
      const int t = t0 + r + 8 * half;
      float v = acc[r] + bias;
      v = v > 0.f ? v : 0.01f * v;
      h1[((size_t)b * Tn + t) * 160 + n] = v;
    }
  }
}

// ---------------------------------------------------------------------------
// Mel conv2+conv3 fused: h2 = leaky(conv1d(h1,K=3,160->80)); emb = conv1d(h2,K=1)
// ---------------------------------------------------------------------------
__global__ void k_mel_conv23(const float* __restrict__ h1,
                             const float* __restrict__ W2m, const float* __restrict__ mb2,
                             const float* __restrict__ mw3, const float* __restrict__ mb3,
                             const int* __restrict__ mel_len,
                             float* __restrict__ mel_emb) {
  __shared__ float h2[16 * 80];
  const int b = blockIdx.y, t0 = blockIdx.x * 16;
  const int tid = threadIdx.x, w = tid >> 5, lane = tid & 31;
  const int m = lane & 15, half = lane >> 4;
  const float* xb = h1 + (size_t)b * Tn * 160;

  if (w < 5) {  // Stage 1: h2(16x80) = leaky(im2col(16x480) @ W2'(480x80) + b2)
    const int n = w * 16 + m;
    v8f acc = {};
    for (int kb = 0; kb < 480; kb += 4) {
      const int col = kb + 2 * half;
      const int so = (col >= 320) ? 2 : ((col >= 160) ? 1 : 0);
      const int ci = col - so * 160;
      const int seq = t0 + m - 1 + so;
      v2f a = {};
      if (seq >= 0 && seq < Tn) a = *(const v2f*)(xb + (size_t)seq * 160 + ci);
      v2f bf = *(const v2f*)(W2m + (size_t)(col >> 1) * 160 + n * 2);
      acc = wmma4(a, bf, acc);
    }
    const float bias = mb2[n];
    for (int r = 0; r < 8; ++r) {
      const int mm = r + 8 * half;
      float v = acc[r] + bias;
      h2[mm * 80 + n] = v > 0.f ? v : 0.01f * v;
    }
  }
  __syncthreads();

  if (w < 5) {  // Stage 2: emb(16x80) = h2 @ W3^T(80x80) + b3, masked by mel_len
    const int co = w * 16 + m;
    v8f acc = {};
    for (int kb = 0; kb < 80; kb += 4) {
      v2f a = *(const v2f*)(&h2[m * 80 + kb + 2 * half]);
      v2f bf = *(const v2f*)(mw3 + (size_t)co * 80 + kb + 2 * half);
      acc = wmma4(a, bf, acc);
    }
    const float bias = mb3[co];
    const int mlen = mel_len[b];
    for (int r = 0; r < 8; ++r) {
      const int t = t0 + r + 8 * half;
      const float v = (t < mlen) ? acc[r] + bias : 0.f;
      mel_emb[((size_t)b * Tn + t) * DMEL + co] = v;
    }
  }
}

// ---------------------------------------------------------------------------
// Row squared-norms of both embeddings.
// ---------------------------------------------------------------------------
__global__ void k_norms(const float* __restrict__ te, const float* __restrict__ me,
                        float* __restrict__ na, float* __restrict__ nb) {
  const int i = blockIdx.x * blockDim.x + threadIdx.x;
  const int nA = Bn * Sn;
  if (i < nA) {
    const float* p = te + (size_t)i * DMEL;
    float s = 0.f;
    for (int j = 0; j < DMEL; ++j) s += p[j] * p[j];
    na[i] = s;
  } else if (i < nA + Bn * Tn) {
    const int r = i - nA;
    const float* p = me + (size_t)r * DMEL;
    float s = 0.f;
    for (int j = 0; j < DMEL; ++j) s += p[j] * p[j];
    nb[r] = s;
  }
}

// ---------------------------------------------------------------------------
// cdist GEMM (WMMA): logits = valid ? -sqrt(max(|a|^2+|b|^2-2ab,1e-12)) : -1e9
// Tiles staged to LDS with GLOBAL_LOAD_ASYNC_TO_LDS_B128 (ASYNCcnt), then
// WMMA fragments come from ds_load. Block = 8 waves, 32(S) x 64(T) tile.
// ---------------------------------------------------------------------------
__global__ void k_cdist(const float* __restrict__ te, const float* __restrict__ me,
                        const float* __restrict__ na, const float* __restrict__ nb,
                        const int* __restrict__ src_len, const int* __restrict__ mel_len,
                        float* __restrict__ softA) {
  __shared__ float ste[32 * DMEL];  // 10 KB
  __shared__ float sme[64 * DMEL];  // 20 KB
  const int b = blockIdx.z;
  const int tid = threadIdx.x, w = tid >> 5, lane = tid & 31;
  const int m = lane & 15, half = lane >> 4;

  const unsigned te_goff = (unsigned)((((size_t)b * Sn + blockIdx.y * 32) * DMEL) * 4);
  const unsigned me_goff = (unsigned)((((size_t)b * Tn + blockIdx.x * 64) * DMEL) * 4);
  for (int i = tid; i < (32 * DMEL) / 4; i += 256) {  // 640 x 16B
    async_g2l_b128((unsigned)(uintptr_t)(&ste[i * 4]), te_goff + i * 16u, te);
  }
  for (int i = tid; i < (64 * DMEL) / 4; i += 256) {  // 1280 x 16B
    async_g2l_b128((unsigned)(uintptr_t)(&sme[i * 4]), me_goff + i * 16u, me);
  }
  wait_asynccnt0();
  __syncthreads();

  const float* ta = &ste[(w >> 2) * 16 * DMEL];
  const float* tb = &sme[(w & 3) * 16 * DMEL];
  v8f acc = {};
  for (int kb = 0; kb < 80; kb += 4) {
    v2f a = *(const v2f*)(ta + m * DMEL + kb + 2 * half);
    v2f bf = *(const v2f*)(tb + m * DMEL + kb + 2 * half);
    acc = wmma4(a, bf, acc);
  }
  const int s0 = blockIdx.y * 32 + 16 * (w >> 2);
  const int t0 = blockIdx.x * 64 + 16 * (w & 3);
  const int slen = src_len[b], mlen = mel_len[b];
  const int t = t0 + m;
  const float nbv = nb[b * Tn + t];
  const bool tv = t < mlen;
  for (int r = 0; r < 8; ++r) {
    const int s = s0 + r + 8 * half;
    const float d2 = na[b * Sn + s] + nbv - 2.f * acc[r];
    const float dist = sqrtf(fmaxf(d2, 1e-12f));
    softA[((size_t)b * Sn + s) * Tn + t] = (tv && s < slen) ? -dist : -1e9f;
  }
}

// ---------------------------------------------------------------------------
// Column softmax over S (axis=1), in place, zeros outside mask.
// ---------------------------------------------------------------------------
__global__ void k_softmax(float* __restrict__ softA, const int* __restrict__ src_len,
                          const int* __restrict__ mel_len) {
  const int col = blockIdx.x * blockDim.x + threadIdx.x;  // over B*T
  const int b = col >> 11, t = col & (Tn - 1);
  float* base = softA + (size_t)b * Sn * Tn + t;
  const int slen = src_len[b];
  if (t >= mel_len[b]) {
    for (int s = 0; s < Sn; ++s) base[(size_t)s * Tn] = 0.f;
    return;
  }
  float mx = -INFINITY;
  for (int s = 0; s < slen; ++s) {
    if (s + 16 < slen) __builtin_prefetch(base + (size_t)(s + 16) * Tn, 0, 0);
    mx = fmaxf(mx, base[(size_t)s * Tn]);
  }
  float sum = 0.f;
  for (int s = 0; s < slen; ++s) sum += __expf(base[(size_t)s * Tn] - mx);
  const float inv = 1.f / sum;
  for (int s = 0; s < slen; ++s) base[(size_t)s * Tn] = __expf(base[(size_t)s * Tn] - mx) * inv;
  for (int s = slen; s < Sn; ++s) base[(size_t)s * Tn] = 0.f;
}

// ---------------------------------------------------------------------------
// hard_A = 0 everywhere; rev = !(s<slen && t<mlen)
// ---------------------------------------------------------------------------
__global__ void k_fill(float* __restrict__ hardA, float* __restrict__ rev,
                       const int* __restrict__ src_len, const int* __restrict__ mel_len) {
  const size_t idx = (size_t)blockIdx.x * blockDim.x + threadIdx.x;
  if (idx >= (size_t)Bn * Sn * Tn) return;
  const int t = (int)(idx & (Tn - 1));
  const size_t r = idx >> 11;
  const int s = (int)(r & (Sn - 1));
  const int b = (int)(r >> 9);
  hardA[idx] = 0.f;
  rev[idx] = (s < src_len[b] && t < mel_len[b]) ? 0.f : 1.f;
}

// ---------------------------------------------------------------------------
// Monotonic alignment forward DP. One workgroup (16 waves) per batch row,
// double-buffered LDS, wave32 ballot packs the direction bits.
// ---------------------------------------------------------------------------
__global__ void k_mas_fwd(const float* __restrict__ softA, const int* __restrict__ src_len,
                          const int* __restrict__ mel_len, unsigned* __restrict__ dirs) {
  __shared__ float vbuf[2][Sn];
  const int b = blockIdx.x, s = threadIdx.x;
  const int slen = src_len[b], mlen = mel_len[b];
  const float* colbase = softA + (size_t)b * Sn * Tn + (size_t)s * Tn;
  const bool srow = s < slen;
  vbuf[0][s] = 0.f;
  __syncthreads();
  int cur = 0;
  for (int t = 0; t < Tn; ++t) {
    const float v = vbuf[cur][s];
    const float v0 = (s > 0) ? vbuf[cur][s - 1] : -INFINITY;
    const bool keep = v >= v0;
    const float vmax = keep ? v : v0;
    const float colv = (srow && t < mlen) ? colbase[t] : 0.f;
    const float vnew = (s <= t) ? vmax + colv : -INFINITY;
    const unsigned word = __builtin_amdgcn_ballot_w32(keep);
    if ((s & 31) == 0) dirs[(((size_t)b * Tn + t) << 4) + (s >> 5)] = word;
    vbuf[cur ^ 1][s] = vnew;
    __syncthreads();
    cur ^= 1;
  }
}

// ---------------------------------------------------------------------------
// Backtrack: one lane per batch walks T-1..0 writing the one-hot path.
// ---------------------------------------------------------------------------
__global__ void k_mas_bwd(const unsigned* __restrict__ dirs, const int* __restrict__ src_len,
                          const int* __restrict__ mel_len, float* __restrict__ hardA) {
  const int b = blockIdx.x;
  if (threadIdx.x != 0) return;
  int idx = src_len[b] - 1;
  const int mlen = mel_len[b];
  for (int t = Tn - 1; t >= 0; --t) {
    int d = 1;
    if (t < mlen) {
      hardA[((size_t)b * Sn + idx) * Tn + t] = 1.f;
      const unsigned word = dirs[(((size_t)b * Tn + t) << 4) + (idx >> 5)];
      d = (int)((word >> (idx & 31)) & 1u);
    }
    idx += d - 1;
    if (idx < 0) idx = 0;
  }
}

extern "C" void kernel_launch(void* const* d_in, const int* in_sizes, int n_in,
                              void* d_out, int out_size, void* d_ws, size_t ws_size,
                              hipStream_t stream) {
  (void)in_sizes; (void)n_in; (void)out_size; (void)ws_size;
  const float* text = (const float*)d_in[0];
  const float* mel = (const float*)d_in[1];
  const int* src_len = (const int*)d_in[2];
  const int* mel_len = (const int*)d_in[3];
  // d_in[4..6]: src_mask / mel_mask / attn_mask (derived from lengths instead)
  const float* tw1 = (const float*)d_in[7];
  const float* tb1 = (const float*)d_in[8];
  const float* tw2 = (const float*)d_in[9];
  const float* tb2 = (const float*)d_in[10];
  const float* mw1 = (const float*)d_in[11];
  const float* mb1 = (const float*)d_in[12];
  const float* mw2 = (const float*)d_in[13];
  const float* mb2 = (const float*)d_in[14];
  const float* mw3 = (const float*)d_in[15];
  const float* mb3 = (const float*)d_in[16];

  float* softA = (float*)d_out;
  float* hardA = softA + (size_t)Bn * Sn * Tn;
  float* rev = hardA + (size_t)Bn * Sn * Tn;

  float* wsf = (float*)d_ws;
  float* text_emb = wsf;                                  //  5.2 MB
  float* mel_h1 = text_emb + (size_t)Bn * Sn * DMEL;      // 41.9 MB
  float* mel_emb = mel_h1 + (size_t)Bn * Tn * 160;        // 21.0 MB
  float* na = mel_emb + (size_t)Bn * Tn * DMEL;
  float* nb = na + (size_t)Bn * Sn;
  unsigned* dirs = (unsigned*)(nb + (size_t)Bn * Tn);     //  4.2 MB
  float* W1p = (float*)(dirs + (size_t)Bn * Tn * (Sn / 32));  // 384*512*2
  float* W1m = W1p + (size_t)384 * 512 * 2;                   // 120*160*2
  float* W2m = W1m + (size_t)120 * 160 * 2;                   // 240*80*2

  const int repack_elems = 384 * 512 + 120 * 160 + 240 * 80;
  k_repack<<<(repack_elems + 255) / 256, 256, 0, stream>>>(tw1, mw1, mw2, W1p, W1m, W2m);
  k_text_enc<<<dim3(Sn / 16, Bn), 256, 0, stream>>>(text, W1p, tb1, tw2, tb2, src_len, text_emb);
  k_mel_conv1<<<dim3(Tn / 16, Bn), 256, 0, stream>>>(mel, W1m, mb1, mel_h1);
  k_mel_conv23<<<dim3(Tn / 16, Bn), 256, 0, stream>>>(mel_h1, W2m, mb2, mw3, mb3, mel_len, mel_emb);
  k_norms<<<(Bn * Sn + Bn * Tn + 255) / 256, 256, 0, stream>>>(text_emb, mel_emb, na, nb);
  k_cdist<<<dim3(Tn / 64, Sn / 32, Bn), 256, 0, stream>>>(text_emb, mel_emb, na, nb,
                                                          src_len, mel_len, softA);
  k_softmax<<<(Bn * Tn) / 256, 256, 0, stream>>>(softA, src_len, mel_len);
  k_fill<<<(int)(((size_t)Bn * Sn * Tn + 255) / 256), 256, 0, stream>>>(hardA, rev, src_len, mel_len);
  k_mas_fwd<<<Bn, Sn, 0, stream>>>(softA, src_len, mel_len, dirs);
  k_mas_bwd<<<Bn, 32, 0, stream>>>(dirs, src_len, mel_len, hardA);
}